// SimpleMHSA_16372415332876
// MI455X (gfx1250) — compile-verified
//
#include <hip/hip_runtime.h>

// SimpleMHSA on MI455X (gfx1250, wave32).
// B=4, C=256, H=W=48 -> N=2304, nh=4, hd=64, fp32 throughout.
// Pipeline: [qkv GEMM + scatter] -> [flash attention w/ async LDS staging]
//           -> [proj GEMM].  All matmuls: V_WMMA_F32_16X16X4_F32.

typedef float v2f __attribute__((ext_vector_type(2)));
typedef float v8f __attribute__((ext_vector_type(8)));

#define B_     4
#define C_     256
#define NH_    4
#define HD_    64
#define N_     2304
#define NT_    144       // N_/16 key tiles
#define KV_TILE 1024     // 16x64 floats = 4KB, contiguous in workspace
#define SCALE_ 0.0625f   // C^-0.5

__device__ __forceinline__ v8f wmma_f32(v2f a, v2f b, v8f c) {
  return __builtin_amdgcn_wmma_f32_16x16x4_f32(false, a, false, b, (short)0, c,
                                               false, false);
}

// --- CDNA5 async global->LDS staging (ASYNCcnt) ----------------------------
// One 16x64 fp32 tile = 4KB contiguous; 8 x b128 per-lane async loads
// (32 lanes x 16B = 512B per instruction). INST_OFFSET applies to both the
// LDS and global addresses (CDNA5 ISA).
#define ASYNC_LD128(ldsa, gpa, OFF)                                           \
  asm volatile("global_load_async_to_lds_b128 %0, %1, off offset:" #OFF       \
               ::"v"(ldsa), "v"(gpa) : "memory")

__device__ __forceinline__ void stage_tile(float* dst_lds, const float* src,
                                           int lane) {
  unsigned ldsa = (unsigned)(unsigned long long)(const void*)dst_lds +
                  (unsigned)lane * 16u;
  const char* gpa = (const char*)src + lane * 16;
  ASYNC_LD128(ldsa, gpa, 0);
  ASYNC_LD128(ldsa, gpa, 512);
  ASYNC_LD128(ldsa, gpa, 1024);
  ASYNC_LD128(ldsa, gpa, 1536);
  ASYNC_LD128(ldsa, gpa, 2048);
  ASYNC_LD128(ldsa, gpa, 2560);
  ASYNC_LD128(ldsa, gpa, 3072);
  ASYNC_LD128(ldsa, gpa, 3584);
}

__device__ __forceinline__ void wait_async_le(void) {  // prev tile landed
  asm volatile("s_wait_asynccnt 0x10" ::: "memory");
}
__device__ __forceinline__ void wait_async_all(void) {
  asm volatile("s_wait_asynccnt 0x0" ::: "memory");
}
__device__ __forceinline__ void wait_ds0(void) {
  asm volatile("s_wait_dscnt 0x0" ::: "memory");
}

// ---------------------------------------------------------------------------
// Kernel A: qkv[b,o,n] = sum_c w_qkv[o,c] * x[b,c,n] + b_qkv[o]
// One wave = 16x32 output tile (two 16x16 accs share each A fragment).
// Scattered into Q/K/V workspaces laid out [b*nh+h][n][hd].
// ---------------------------------------------------------------------------
__global__ __launch_bounds__(256) void qkv_proj_kernel(
    const float* __restrict__ x, const float* __restrict__ w_qkv,
    const float* __restrict__ b_qkv, float* __restrict__ qws,
    float* __restrict__ kws, float* __restrict__ vws) {
  const int lane = threadIdx.x & 31;
  const int wave = threadIdx.x >> 5;
  const int w    = blockIdx.x * 8 + wave;
  const int b    = w / (48 * 72);
  const int rem  = w % (48 * 72);
  const int o0   = (rem / 72) * 16;
  const int n0   = (rem % 72) * 32;

  const int lo = lane & 15;
  const int hi = lane >> 4;
  const int kk = hi * 2;

  const float* xb = x + (size_t)b * C_ * N_;
  v8f acc0 = {}, acc1 = {};
#pragma unroll 4
  for (int k0 = 0; k0 < C_; k0 += 4) {
    v2f a = *(const v2f*)(w_qkv + (size_t)(o0 + lo) * C_ + k0 + kk);
    const float* xr0 = xb + (size_t)(k0 + kk) * N_;
    const float* xr1 = xr0 + N_;
    v2f b0, b1;
    b0[0] = xr0[n0 + lo];       b0[1] = xr1[n0 + lo];
    b1[0] = xr0[n0 + 16 + lo];  b1[1] = xr1[n0 + 16 + lo];
    acc0 = wmma_f32(a, b0, acc0);
    acc1 = wmma_f32(a, b1, acc1);
  }
#pragma unroll
  for (int r = 0; r < 8; ++r) {
    const int o    = o0 + r + 8 * hi;
    const float bv = b_qkv[o];
    const int seg  = o >> 8;      // 0=q 1=k 2=v (uniform over wave)
    const int c2   = o & 255;
    const int head = c2 >> 6;     // uniform over wave
    const int d    = c2 & 63;
    float* dst = (seg == 0) ? qws : (seg == 1) ? kws : vws;
    float* base = dst + (size_t)(b * NH_ + head) * N_ * HD_ + d;
    base[(size_t)(n0 + lo) * HD_]      = acc0[r] + bv;
    base[(size_t)(n0 + 16 + lo) * HD_] = acc1[r] + bv;
  }
}

// ---------------------------------------------------------------------------
// Kernel B: flash attention, one wave = 16 query rows for one (b,head).
// K/V tiles double-buffered in LDS via async global->LDS loads; S = Q K^T
// via 16 WMMAs (B frags = ds_load_b64), online softmax with shfl_xor row
// reductions, P relayered D->A through LDS, O += P V via 16 WMMAs.
// ---------------------------------------------------------------------------
__global__ __launch_bounds__(64) void attn_kernel(
    const float* __restrict__ qws, const float* __restrict__ kws,
    const float* __restrict__ vws, float* __restrict__ ows) {
  __shared__ float kbuf[2][2][KV_TILE];  // [wave][buf][16x64]
  __shared__ float vbuf[2][2][KV_TILE];
  __shared__ float pts[2][16][17];       // per-wave P tile, padded
  const int lane = threadIdx.x & 31;
  const int wave = threadIdx.x >> 5;
  const int w    = blockIdx.x * 2 + wave;
  const int bh   = w / NT_;
  const int q0   = (w % NT_) * 16;
  const int lo = lane & 15;
  const int hi = lane >> 4;
  const int kk = hi * 2;

  const float* qb = qws + ((size_t)bh * N_ + q0) * HD_;
  const float* kb = kws + (size_t)bh * N_ * HD_;
  const float* vb = vws + (size_t)bh * N_ * HD_;

  // Preload all Q A-fragments (16 k-steps over hd=64).
  v2f qa[16];
#pragma unroll
  for (int ks = 0; ks < 16; ++ks)
    qa[ks] = *(const v2f*)(qb + (size_t)lo * HD_ + ks * 4 + kk);

  float m8[8], l8[8];
  v8f oacc[4];
  v8f vzero = {};
#pragma unroll
  for (int r = 0; r < 8; ++r) { m8[r] = -1e30f; l8[r] = 0.0f; }
#pragma unroll
  for (int t = 0; t < 4; ++t) oacc[t] = vzero;

  // Prologue: stage tile 0 into buffer 0.
  stage_tile(&kbuf[wave][0][0], kb, lane);
  stage_tile(&vbuf[wave][0][0], vb, lane);

  for (int j = 0; j < NT_; ++j) {
    const int cur = j & 1;
    if (j + 1 < NT_) {
      wait_ds0();  // LDS reads of buffer cur^1 (iter j-1) fully drained
      stage_tile(&kbuf[wave][cur ^ 1][0], kb + (size_t)(j + 1) * KV_TILE, lane);
      stage_tile(&vbuf[wave][cur ^ 1][0], vb + (size_t)(j + 1) * KV_TILE, lane);
      wait_async_le();   // <=16 outstanding: tile j's 16 loads completed
    } else {
      wait_async_all();
    }
    const float* kt = &kbuf[wave][cur][0];
    const float* vt = &vbuf[wave][cur][0];

    // S tile = Q Kt^T (B frags: aligned ds_load_b64)
    v8f s = vzero;
#pragma unroll
    for (int ks = 0; ks < 16; ++ks) {
      v2f bb = *(const v2f*)(kt + lo * HD_ + ks * 4 + kk);
      s = wmma_f32(qa[ks], bb, s);
    }

    // Online softmax: D-layout VGPR r = one row across a 16-lane group.
    float p[8];
#pragma unroll
    for (int r = 0; r < 8; ++r) {
      float sv = s[r] * SCALE_;
      float t0 = sv;
      t0 = fmaxf(t0, __shfl_xor(t0, 1, 32));
      t0 = fmaxf(t0, __shfl_xor(t0, 2, 32));
      t0 = fmaxf(t0, __shfl_xor(t0, 4, 32));
      t0 = fmaxf(t0, __shfl_xor(t0, 8, 32));
      const float mnew  = fmaxf(m8[r], t0);
      const float alpha = __expf(m8[r] - mnew);
      const float pv    = __expf(sv - mnew);
      float ssum = pv;
      ssum += __shfl_xor(ssum, 1, 32);
      ssum += __shfl_xor(ssum, 2, 32);
      ssum += __shfl_xor(ssum, 4, 32);
      ssum += __shfl_xor(ssum, 8, 32);
      l8[r] = l8[r] * alpha + ssum;
      m8[r] = mnew;
      p[r]  = pv;
#pragma unroll
      for (int t = 0; t < 4; ++t) oacc[t][r] *= alpha;
    }

    // D-layout -> A-layout relayering through LDS (wave-private tile;
    // same-wave LDS ops are in-order, wait_ds0 + clobber for safety).
#pragma unroll
    for (int r = 0; r < 8; ++r) pts[wave][r + 8 * hi][lo] = p[r];
    wait_ds0();
    v2f pa[4];
#pragma unroll
    for (int ks = 0; ks < 4; ++ks) {
      pa[ks][0] = pts[wave][lo][ks * 4 + kk];
      pa[ks][1] = pts[wave][lo][ks * 4 + kk + 1];
    }

    // O(16x64) += P(16x16) @ V(16x64)
#pragma unroll
    for (int t = 0; t < 4; ++t) {
#pragma unroll
      for (int ks = 0; ks < 4; ++ks) {
        const int krow = ks * 4 + kk;
        v2f bb;
        bb[0] = vt[krow * HD_ + t * 16 + lo];
        bb[1] = vt[(krow + 1) * HD_ + t * 16 + lo];
        oacc[t] = wmma_f32(pa[ks], bb, oacc[t]);
      }
    }
  }

  float inv[8];
#pragma unroll
  for (int r = 0; r < 8; ++r) inv[r] = 1.0f / l8[r];
  float* ob = ows + ((size_t)bh * N_ + q0) * HD_;
#pragma unroll
  for (int t = 0; t < 4; ++t)
#pragma unroll
    for (int r = 0; r < 8; ++r)
      ob[(size_t)(r + 8 * hi) * HD_ + t * 16 + lo] = oacc[t][r] * inv[r];
}

// ---------------------------------------------------------------------------
// Kernel C: out[b,o,n] = sum_c w_proj[o,c] * attn_out[b,c,n] + b_proj[o]
// 16x32 tiles; attn_out read from [b*nh+h][n][hd] workspace (consecutive c
// within a head are consecutive floats -> aligned float2 B fragments).
// ---------------------------------------------------------------------------
__global__ __launch_bounds__(256) void proj_kernel(
    const float* __restrict__ ows, const float* __restrict__ w_proj,
    const float* __restrict__ b_proj, float* __restrict__ out) {
  const int lane = threadIdx.x & 31;
  const int wave = threadIdx.x >> 5;
  const int w    = blockIdx.x * 8 + wave;
  const int b    = w / (16 * 72);
  const int rem  = w % (16 * 72);
  const int o0   = (rem / 72) * 16;
  const int n0   = (rem % 72) * 32;
  const int lo = lane & 15;
  const int hi = lane >> 4;
  const int kk = hi * 2;

  const float* xb = ows + (size_t)b * NH_ * N_ * HD_;
  v8f acc0 = {}, acc1 = {};
#pragma unroll 4
  for (int k0 = 0; k0 < C_; k0 += 4) {
    v2f a = *(const v2f*)(w_proj + (size_t)(o0 + lo) * C_ + k0 + kk);
    const int c    = k0 + kk;   // even; (c, c+1) stay within one head
    const int head = c >> 6;
    const int d    = c & 63;
    const float* hb = xb + (size_t)head * N_ * HD_ + d;
    v2f b0 = *(const v2f*)(hb + (size_t)(n0 + lo) * HD_);
    v2f b1 = *(const v2f*)(hb + (size_t)(n0 + 16 + lo) * HD_);
    acc0 = wmma_f32(a, b0, acc0);
    acc1 = wmma_f32(a, b1, acc1);
  }
#pragma unroll
  for (int r = 0; r < 8; ++r) {
    const int o    = o0 + r + 8 * hi;
    const float bv = b_proj[o];
    float* orow = out + (size_t)b * C_ * N_ + (size_t)o * N_;
    orow[n0 + lo]      = acc0[r] + bv;
    orow[n0 + 16 + lo] = acc1[r] + bv;
  }
}

// ---------------------------------------------------------------------------
extern "C" void kernel_launch(void* const* d_in, const int* in_sizes, int n_in,
                              void* d_out, int out_size, void* d_ws,
                              size_t ws_size, hipStream_t stream) {
  const float* x      = (const float*)d_in[0];
  const float* w_qkv  = (const float*)d_in[1];
  const float* b_qkv  = (const float*)d_in[2];
  const float* w_proj = (const float*)d_in[3];
  const float* b_proj = (const float*)d_in[4];
  // d_in[5] = num_heads (scalar) — fixed to 4 by the reference shapes.

  // Workspace: Q, K, V, O each B*nh*N*hd floats = 9.44 MB -> 37.7 MB total.
  float* ws = (float*)d_ws;
  const size_t seg = (size_t)B_ * NH_ * N_ * HD_;
  float* qws = ws;
  float* kws = ws + seg;
  float* vws = ws + 2 * seg;
  float* ows = ws + 3 * seg;

  // Kernel A: 4 * 48 * 72 = 13824 waves, 8 waves/block.
  qkv_proj_kernel<<<13824 / 8, 256, 0, stream>>>(x, w_qkv, b_qkv, qws, kws,
                                                 vws);
  // Kernel B: 16 * 144 = 2304 waves, 2 waves/block (34KB LDS each).
  attn_kernel<<<2304 / 2, 64, 0, stream>>>(qws, kws, vws, ows);
  // Kernel C: 4 * 16 * 72 = 4608 waves, 8 waves/block.
  proj_kernel<<<4608 / 8, 256, 0, stream>>>(ows, w_proj, b_proj,
                                            (float*)d_out);
}